// QuantumAutoencoder_65481071400231
// MI455X (gfx1250) — compile-verified
//
#include <hip/hip_runtime.h>
#include <hip/hip_bf16.h>

#define NQ      12
#define DIM     4096        // 2^12 amplitudes
#define THREADS 256
#define NREPS   3

typedef __attribute__((ext_vector_type(2))) float v2f;
typedef __attribute__((ext_vector_type(8))) float v8f;

// ---------------------------------------------------------------------------
// Kernel 1: per-sample state-vector simulation, state resident in LDS.
//   - Embedding RX layer folded analytically into a product state.
//   - RY (real) + CNOT (permutation) act on float2 (Re,Im) in LDS.
//   - <Z_i> via wave32 shuffle reduction + LDS float atomics.
// One block per batch sample; 256 threads (8 waves) per block; 32KB LDS state.
// ---------------------------------------------------------------------------
__global__ __launch_bounds__(THREADS) void qsim_kernel(
    const float* __restrict__ x,      // [256][12] embedding angles
    const float* __restrict__ qp,     // [72] ansatz params (REPS,2,NQ)
    float* __restrict__ z)            // [256][12] PauliZ expvals (out)
{
  __shared__ float2 S[DIM];           // state: (Re, Im)
  __shared__ float  cv[NQ], sv[NQ];   // embedding cos/sin
  __shared__ float  zacc[NQ];

  const int tid = threadIdx.x;
  const int b   = blockIdx.x;

  if (tid < NQ) {
    float h = 0.5f * x[b * NQ + tid];
    cv[tid] = __builtin_cosf(h);
    sv[tid] = __builtin_sinf(h);
    zacc[tid] = 0.0f;
  }
  __syncthreads();

  // --- initial product state: amp(idx) = (-i)^popc(idx) * prod(c or s) -----
  for (int j = 0; j < DIM / THREADS; ++j) {
    int idx = tid + j * THREADS;
    float mag = 1.0f;
#pragma unroll
    for (int w = 0; w < NQ; ++w) {
      int bit = (idx >> (NQ - 1 - w)) & 1;
      mag *= bit ? sv[w] : cv[w];
    }
    int k = __popc(idx) & 3;          // phase (-i)^k
    float2 a;
    a.x = (k == 0) ?  mag : ((k == 2) ? -mag : 0.0f);
    a.y = (k == 1) ? -mag : ((k == 3) ?  mag : 0.0f);
    S[idx] = a;
  }
  __syncthreads();

  // --- ansatz: REPS x 2 x (RY layer + CNOT chain). All-real operators. -----
  for (int r = 0; r < NREPS; ++r) {
    for (int l = 0; l < 2; ++l) {
      // RY layer (shared params -> scalar loads, uniform c/s)
      for (int w = 0; w < NQ; ++w) {
        float th = 0.5f * qp[(r * 2 + l) * NQ + w];
        float c = __builtin_cosf(th);
        float s = __builtin_sinf(th);
        const int bpos = NQ - 1 - w;
        const int mask = (1 << bpos) - 1;
#pragma unroll
        for (int j = 0; j < (DIM / 2) / THREADS; ++j) {   // 8 pairs/thread
          int pi = tid + j * THREADS;
          int i0 = ((pi & ~mask) << 1) | (pi & mask);
          int i1 = i0 | (1 << bpos);
          float2 a0 = S[i0];
          float2 a1 = S[i1];
          float2 n0, n1;
          n0.x = c * a0.x - s * a1.x;  n0.y = c * a0.y - s * a1.y;
          n1.x = s * a0.x + c * a1.x;  n1.y = s * a0.y + c * a1.y;
          S[i0] = n0;
          S[i1] = n1;
        }
        __syncthreads();
      }
      // CNOT chain: ctrl=i, tgt=i+1 -> pure LDS swap on ctrl-bit==1 subspace
      for (int i = 0; i < NQ - 1; ++i) {
        const int bt    = NQ - 2 - i;          // target bit position
        const int maskt = (1 << bt) - 1;
#pragma unroll
        for (int j = 0; j < (DIM / 4) / THREADS; ++j) {   // 4 swaps/thread
          int pi = tid + j * THREADS;
          int i0 = ((pi & ~maskt) << 2) | (1 << (bt + 1)) | (pi & maskt);
          int i1 = i0 | (1 << bt);
          float2 t0 = S[i0];
          float2 t1 = S[i1];
          S[i0] = t1;
          S[i1] = t0;
        }
        __syncthreads();
      }
    }
  }

  // --- <Z_w> = sum_idx |amp|^2 * (1 - 2*bit_w) ------------------------------
  float part[NQ];
#pragma unroll
  for (int w = 0; w < NQ; ++w) part[w] = 0.0f;

  for (int j = 0; j < DIM / THREADS; ++j) {
    int idx = tid + j * THREADS;
    float2 a = S[idx];
    float pr = a.x * a.x + a.y * a.y;
#pragma unroll
    for (int w = 0; w < NQ; ++w)
      part[w] += ((idx >> (NQ - 1 - w)) & 1) ? -pr : pr;
  }

  // wave32 xor-tree reduction, then one LDS atomic per wave per wire
#pragma unroll
  for (int w = 0; w < NQ; ++w) {
    float v = part[w];
#pragma unroll
    for (int off = 16; off > 0; off >>= 1)
      v += __shfl_xor(v, off, 32);
    part[w] = v;
  }
  if ((tid & 31) == 0) {
#pragma unroll
    for (int w = 0; w < NQ; ++w)
      atomicAdd(&zacc[w], part[w]);
  }
  __syncthreads();

  if (tid < NQ) z[b * NQ + tid] = zacc[tid];
}

// ---------------------------------------------------------------------------
// Kernel 2: out = z @ W^T + b via V_WMMA_F32_16X16X4_F32.
// One wave per 16-row batch tile (16 blocks x 32 threads). K=12 as 3 chained
// K=4 WMMAs. Full EXEC during WMMA; column mask applied only at the store.
// Layouts per ISA 7.12.2 (32-bit 16x4 A / 4x16 B, 16x16 f32 C/D):
//   A: lane L, VGPR v  -> M = L%16, K = k0 + (L<16 ? v : v+2)
//   B: lane L, VGPR v  -> N = L%16, K = k0 + (L<16 ? v : v+2)
//   C/D: lane L, VGPR r -> M = r + (L<16 ? 0 : 8), N = L%16
// ---------------------------------------------------------------------------
__global__ __launch_bounds__(32) void decode_wmma_kernel(
    const float* __restrict__ z,      // [256][12]
    const float* __restrict__ W,      // [12][12] row-major (out_j, in_k)
    const float* __restrict__ bias,   // [12]
    float* __restrict__ out)          // [256][12]
{
  const int lane    = threadIdx.x;
  const int m       = lane & 15;      // col index N / row index M within tile
  const int hi      = lane >> 4;      // lane group: 0 -> K+{0,1}, 1 -> K+{2,3}
  const int rowBase = blockIdx.x * 16;

  v8f acc = {0.f, 0.f, 0.f, 0.f, 0.f, 0.f, 0.f, 0.f};

#pragma unroll
  for (int k0 = 0; k0 < NQ; k0 += 4) {
    const int ka = k0 + (hi ? 2 : 0);
    v2f a, bm;
    // A = z tile: rows rowBase+m, cols ka, ka+1   (ka <= 11 always)
    a.x = z[(rowBase + m) * NQ + ka];
    a.y = z[(rowBase + m) * NQ + ka + 1];
    // B[k][n] = W[n][k] so that D[m][n] = sum_k z[m][k] * W[n][k]
    bm.x = (m < NQ) ? W[m * NQ + ka]     : 0.0f;
    bm.y = (m < NQ) ? W[m * NQ + ka + 1] : 0.0f;
    acc = __builtin_amdgcn_wmma_f32_16x16x4_f32(
        /*neg_a=*/false, a, /*neg_b=*/false, bm,
        /*c_mod=*/(short)0, acc, /*reuse_a=*/false, /*reuse_b=*/false);
  }

  if (m < NQ) {                       // only 12 valid output columns
    const float bv = bias[m];
#pragma unroll
    for (int r = 0; r < 8; ++r) {
      const int row = rowBase + r + hi * 8;
      out[row * NQ + m] = acc[r] + bv;
    }
  }
}

// ---------------------------------------------------------------------------
extern "C" void kernel_launch(void* const* d_in, const int* in_sizes, int n_in,
                              void* d_out, int out_size, void* d_ws, size_t ws_size,
                              hipStream_t stream) {
  const float* x  = (const float*)d_in[0];   // [256*12]
  const float* qp = (const float*)d_in[1];   // [72]
  const float* W  = (const float*)d_in[2];   // [12*12]
  const float* bi = (const float*)d_in[3];   // [12]
  float* zbuf = (float*)d_ws;                // scratch: 256*12 floats
  float* out  = (float*)d_out;               // [256*12]

  qsim_kernel<<<256, THREADS, 0, stream>>>(x, qp, zbuf);
  decode_wmma_kernel<<<16, 32, 0, stream>>>(zbuf, W, bi, out);
}